// GraphAttention_32117765439638
// MI455X (gfx1250) — compile-verified
//
#include <hip/hip_runtime.h>
#include <math.h>

#define DIN  1024
#define DOUT 1024
#define NEG_SLOPE 0.1f
#define EPS 1e-5f

typedef __attribute__((ext_vector_type(2))) float v2f;
typedef __attribute__((ext_vector_type(8))) float v8f;

// ---------------------------------------------------------------------------
// Kernel 0: Wt[col][k] = W[k][col]  (so B fragments become one b64 load),
// and zero the CSR degree counters.
// ---------------------------------------------------------------------------
__global__ void k_wt(const float* __restrict__ W, float* __restrict__ Wt,
                     int* __restrict__ count, int n) {
  int i = blockIdx.x * blockDim.x + threadIdx.x;   // over DIN*DOUT
  if (i < DIN * DOUT) {
    int k = i >> 10, col = i & 1023;
    Wt[(size_t)col * DIN + k] = W[i];
  }
  if (i < n) count[i] = 0;
}

// ---------------------------------------------------------------------------
// Kernel 1: h = x @ W via V_WMMA_F32_16X16X4_F32 (full fp32 precision).
// One block per 16-row tile (N/16 = 625), 8 waves; wave w owns cols
// [w*128, w*128+128) = 8 accumulators. B read from transposed Wt: the
// (K=k0,k0+1) pair per lane is contiguous -> single 8B load.
// ---------------------------------------------------------------------------
__global__ __launch_bounds__(256) void gat_gemm_h(const float* __restrict__ x,
                                                  const float* __restrict__ Wt,
                                                  float* __restrict__ h) {
  const int tileM   = blockIdx.x;
  const int wave    = threadIdx.x >> 5;
  const int lane    = threadIdx.x & 31;
  const int khalf   = (lane >> 4) * 2;          // 0 or 2
  const int colBase = wave * 128;
  const int ncol    = lane & 15;
  const int rowA    = tileM * 16 + (lane & 15);

  v8f acc[8];
#pragma unroll
  for (int t = 0; t < 8; ++t) acc[t] = (v8f){0.f,0.f,0.f,0.f,0.f,0.f,0.f,0.f};

  const float* __restrict__ xrow = x + (size_t)rowA * DIN;

  for (int kk = 0; kk < DIN; kk += 4) {
    const int k0 = kk + khalf;                  // even -> 8B aligned
    v2f afrag = *(const v2f*)(xrow + k0);
#pragma unroll
    for (int t = 0; t < 8; ++t) {
      const int col = colBase + t * 16 + ncol;
      v2f bfrag = *(const v2f*)(Wt + (size_t)col * DIN + k0);
      acc[t] = __builtin_amdgcn_wmma_f32_16x16x4_f32(
          false, afrag, false, bfrag, (short)0, acc[t], false, false);
    }
  }

  const int rowHi = (lane >> 4) * 8;
#pragma unroll
  for (int t = 0; t < 8; ++t) {
    const int col = colBase + t * 16 + ncol;
#pragma unroll
    for (int r = 0; r < 8; ++r)
      h[(size_t)(tileM * 16 + rowHi + r) * DOUT + col] = acc[t][r];
  }
}

// ---------------------------------------------------------------------------
// Kernel 2: al = h @ a[:1024], ar = h @ a[1024:]. Wave-per-row GEMV,
// wave32 butterfly reduction.
// ---------------------------------------------------------------------------
__global__ __launch_bounds__(256) void gat_alar(const float* __restrict__ h,
                                                const float* __restrict__ a,
                                                float* __restrict__ al,
                                                float* __restrict__ ar,
                                                int n) {
  const int row = blockIdx.x * 8 + (threadIdx.x >> 5);
  if (row >= n) return;
  const int lane = threadIdx.x & 31;
  const float* __restrict__ hr = h + (size_t)row * DOUT;
  float sl = 0.f, sr = 0.f;
  for (int c = lane; c < DOUT; c += 32) {
    float v = hr[c];
    sl = fmaf(v, a[c], sl);
    sr = fmaf(v, a[DOUT + c], sr);
  }
#pragma unroll
  for (int m = 16; m >= 1; m >>= 1) {
    sl += __shfl_xor(sl, m, 32);
    sr += __shfl_xor(sr, m, 32);
  }
  if (lane == 0) { al[row] = sl; ar[row] = sr; }
}

// ---------------------------------------------------------------------------
// Kernel 3: per-edge score ev = exp(-leaky_relu(al[src]+ar[dst]))
// and degree histogram (int atomics only).
// ---------------------------------------------------------------------------
__global__ void k_count(const float* __restrict__ al, const float* __restrict__ ar,
                        const int* __restrict__ src, const int* __restrict__ dst,
                        float* __restrict__ ev, int* __restrict__ count, int E) {
  int e = blockIdx.x * blockDim.x + threadIdx.x;
  if (e >= E) return;
  int s = src[e], d = dst[e];
  float sc = al[s] + ar[d];
  float lr = sc > 0.f ? sc : NEG_SLOPE * sc;
  ev[e] = expf(-lr);
  atomicAdd(&count[s], 1);
}

// ---------------------------------------------------------------------------
// Kernel 4: single-block exclusive prefix sum over row degrees (N=10000),
// producing rowstart[N+1] and a fill cursor copy.
// ---------------------------------------------------------------------------
__global__ __launch_bounds__(1024) void k_scan(const int* __restrict__ count,
                                               int* __restrict__ rowstart,
                                               int* __restrict__ cursor, int n) {
  __shared__ int buf[1024];
  __shared__ int carry;
  const int tid = threadIdx.x;
  if (tid == 0) carry = 0;
  __syncthreads();
  for (int base = 0; base < n; base += 1024) {
    int i = base + tid;
    int v = (i < n) ? count[i] : 0;
    buf[tid] = v;
    __syncthreads();
#pragma unroll
    for (int off = 1; off < 1024; off <<= 1) {
      int t = (tid >= off) ? buf[tid - off] : 0;
      __syncthreads();
      buf[tid] += t;
      __syncthreads();
    }
    int excl = buf[tid] - v + carry;
    if (i < n) { rowstart[i] = excl; cursor[i] = excl; }
    int blocktotal = buf[1023];
    __syncthreads();
    if (tid == 0) carry += blocktotal;
    __syncthreads();
  }
  if (tid == 0) rowstart[n] = carry;
}

// ---------------------------------------------------------------------------
// Kernel 5: CSR fill (int atomics on cursors).
// ---------------------------------------------------------------------------
__global__ void k_fill(const int* __restrict__ src, int* __restrict__ cursor,
                       int* __restrict__ eidx, int E) {
  int e = blockIdx.x * blockDim.x + threadIdx.x;
  if (e >= E) return;
  int pos = atomicAdd(&cursor[src[e]], 1);
  eidx[pos] = e;
}

__device__ __forceinline__ float elu1(float v) {
  return v > 0.f ? v : expm1f(v);
}

// ---------------------------------------------------------------------------
// Kernel 6: gather-aggregate. One block per src row; each thread owns 4
// contiguous cols (float4). Accumulates ev-weighted h[dst] rows and the ev
// rowsum entirely in registers (zero fp32 atomics), then fused normalize+ELU
// straight to d_out. Edge list insertion-sorted in LDS for deterministic
// accumulation order across replays.
// ---------------------------------------------------------------------------
__global__ __launch_bounds__(256) void k_row(const float* __restrict__ h,
                                             const float* __restrict__ ev,
                                             const int* __restrict__ eidx,
                                             const int* __restrict__ dst,
                                             const int* __restrict__ rowstart,
                                             float* __restrict__ out) {
  __shared__ int sl[256];
  const int r   = blockIdx.x;
  const int beg = rowstart[r];
  const int deg = rowstart[r + 1] - beg;
  const int c   = threadIdx.x * 4;

  float4 acc = make_float4(0.f, 0.f, 0.f, 0.f);
  float evsum = 0.f;

  if (deg <= 256) {
    if (threadIdx.x < deg) sl[threadIdx.x] = eidx[beg + threadIdx.x];
    __syncthreads();
    if (threadIdx.x == 0) {           // tiny deterministic sort (avg deg ~16)
      for (int i = 1; i < deg; ++i) {
        int key = sl[i], j = i - 1;
        while (j >= 0 && sl[j] > key) { sl[j + 1] = sl[j]; --j; }
        sl[j + 1] = key;
      }
    }
    __syncthreads();
    for (int p = 0; p < deg; ++p) {
      const int e = sl[p];
      const float w = ev[e];
      const float4 hv = *(const float4*)(h + (size_t)dst[e] * DOUT + c);
      evsum += w;
      acc.x = fmaf(w, hv.x, acc.x);
      acc.y = fmaf(w, hv.y, acc.y);
      acc.z = fmaf(w, hv.z, acc.z);
      acc.w = fmaf(w, hv.w, acc.w);
    }
  } else {                            // fallback for pathological degree
    for (int p = 0; p < deg; ++p) {
      const int e = eidx[beg + p];
      const float w = ev[e];
      const float4 hv = *(const float4*)(h + (size_t)dst[e] * DOUT + c);
      evsum += w;
      acc.x = fmaf(w, hv.x, acc.x);
      acc.y = fmaf(w, hv.y, acc.y);
      acc.z = fmaf(w, hv.z, acc.z);
      acc.w = fmaf(w, hv.w, acc.w);
    }
  }

  const float inv = 1.f / (evsum + EPS);
  float4 o;
  o.x = elu1(acc.x * inv);
  o.y = elu1(acc.y * inv);
  o.z = elu1(acc.z * inv);
  o.w = elu1(acc.w * inv);
  *(float4*)(out + (size_t)r * DOUT + c) = o;
}

// ---------------------------------------------------------------------------
extern "C" void kernel_launch(void* const* d_in, const int* in_sizes, int n_in,
                              void* d_out, int out_size, void* d_ws, size_t ws_size,
                              hipStream_t stream) {
  const float* x   = (const float*)d_in[0];
  const float* W   = (const float*)d_in[1];
  const float* a   = (const float*)d_in[2];
  const int*   src = (const int*)d_in[3];
  const int*   dst = (const int*)d_in[4];

  const int N = in_sizes[0] / DIN;   // 10000
  const int E = in_sizes[3];         // 160000

  // Workspace layout: h | Wt | al | ar | ev | count | cursor | rowstart | eidx
  float* ws  = (float*)d_ws;
  float* h   = ws;
  float* Wt  = h  + (size_t)N * DOUT;
  float* al  = Wt + (size_t)DIN * DOUT;
  float* ar  = al + N;
  float* ev  = ar + N;
  int* count    = (int*)(ev + E);
  int* cursor   = count + N;
  int* rowstart = cursor + N;
  int* eidx     = rowstart + (N + 1);

  float* out = (float*)d_out;

  k_wt      <<<(DIN * DOUT + 255) / 256, 256, 0, stream>>>(W, Wt, count, N);
  gat_gemm_h<<<N / 16, 256, 0, stream>>>(x, Wt, h);
  gat_alar  <<<(N + 7) / 8, 256, 0, stream>>>(h, a, al, ar, N);
  k_count   <<<(E + 255) / 256, 256, 0, stream>>>(al, ar, src, dst, ev, count, E);
  k_scan    <<<1, 1024, 0, stream>>>(count, rowstart, cursor, N);
  k_fill    <<<(E + 255) / 256, 256, 0, stream>>>(src, cursor, eidx, E);
  k_row     <<<N, 256, 0, stream>>>(h, ev, eidx, dst, rowstart, out);
}